// MultiQueryAttention_11630771438265
// MI455X (gfx1250) — compile-verified
//
#include <hip/hip_runtime.h>
#include <math.h>

typedef __attribute__((ext_vector_type(2))) float v2f;
typedef __attribute__((ext_vector_type(4))) float v4f;
typedef __attribute__((ext_vector_type(8))) float v8f;
typedef __attribute__((ext_vector_type(4))) int v4i;

#define HIDDEN 2048
#define HEADS  16
#define HDIM   128
#define BATCH  2
#define SEQ    2048
#define AT_WAVES 4

#if __has_builtin(__builtin_amdgcn_global_load_async_to_lds_b128)
#define HAVE_ASYNC_LDS 1
#endif

typedef __attribute__((address_space(1))) v4i gv4i;  // global int4
typedef __attribute__((address_space(3))) v4i lv4i;  // LDS int4

__device__ __forceinline__ void async_copy_b128(const float* g, float* l) {
#if defined(HAVE_ASYNC_LDS)
  // GLOBAL_LOAD_ASYNC_TO_LDS_B128: each lane copies 16 bytes from global address g to LDS address l, tracked by ASYNCcnt
  __builtin_amdgcn_global_load_async_to_lds_b128((gv4i*)g, (lv4i*)l, 0, 0);
#else
  *(v4f*)l = *(const v4f*)g;  // synchronous fallback through VGPRs
#endif
}

__device__ __forceinline__ void wait_async0() {
#if __has_builtin(__builtin_amdgcn_s_wait_asynccnt)
  __builtin_amdgcn_s_wait_asynccnt(0);
#elif defined(HAVE_ASYNC_LDS)
  asm volatile("s_wait_asynccnt 0x0" ::: "memory");
#endif
}

__device__ __forceinline__ v8f wmma_f32_16x16x4(v2f a, v2f b, v8f c) {
  return __builtin_amdgcn_wmma_f32_16x16x4_f32(false, a, false, b, (short)0, c,
                                               false, false);
}

// C[MxN] = A[MxK] @ B[KxN] + bias[N], f32 row-major, compile-time shapes.
// 256 threads = 8 waves; block tile 128x128, BK=16, double-buffered LDS staging
// via async global->LDS copies. Each wave computes a 16x128 strip (8 WMMA accs).
template <int M, int N, int K>
__global__ __launch_bounds__(256) void gemm_bias_wmma(
    const float* __restrict__ A, const float* __restrict__ B,
    const float* __restrict__ bias, float* __restrict__ C) {
  constexpr int BM = 128, BN = 128, BK = 16;
  constexpr int NCHUNK = K / BK;
  __shared__ float ldsA[2][BM * BK];  // [m][k], 8 KB per buffer
  __shared__ float ldsB[2][BK * BN];  // [k][n], 8 KB per buffer

  const int t    = threadIdx.x;
  const int lane = t & 31;
  const int wave = t >> 5;
  const int half = lane >> 4;
  const int l16  = lane & 15;

  constexpr int gx = N / BN;
  const int m0 = (blockIdx.x / gx) * BM;
  const int n0 = (blockIdx.x % gx) * BN;

  auto stage = [&](int buf, int k0) {
    // A block: 128x16 = 512 b128 transfers (2 per thread)
#pragma unroll
    for (int j = 0; j < 2; ++j) {
      const int idx = t + j * 256;
      const int row = idx >> 2, q = (idx & 3) << 2;
      async_copy_b128(A + (size_t)(m0 + row) * K + k0 + q,
                      &ldsA[buf][row * BK + q]);
    }
    // B block: 16x128 = 512 b128 transfers (2 per thread)
#pragma unroll
    for (int j = 0; j < 2; ++j) {
      const int idx = t + j * 256;
      const int row = idx >> 5, q = (idx & 31) << 2;
      async_copy_b128(B + (size_t)(k0 + row) * N + n0 + q,
                      &ldsB[buf][row * BN + q]);
    }
  };

  v8f acc[8] = {};
  stage(0, 0);

  for (int c = 0; c < NCHUNK; ++c) {
    wait_async0();
    __syncthreads();
    if (c + 1 < NCHUNK) stage((c + 1) & 1, (c + 1) * BK);

    const int buf = c & 1;
    const float* As = &ldsA[buf][(wave * 16 + l16) * BK + 2 * half];
    const float* Bs = &ldsB[buf][0];
#pragma unroll
    for (int k4 = 0; k4 < BK; k4 += 4) {
      v2f a = *(const v2f*)(As + k4);  // one ds_load_b64, reused across 8 tiles
#pragma unroll
      for (int ns = 0; ns < 8; ++ns) {
        v2f b;
        b.x = Bs[(k4 + 2 * half) * BN + ns * 16 + l16];
        b.y = Bs[(k4 + 2 * half + 1) * BN + ns * 16 + l16];
        acc[ns] = wmma_f32_16x16x4(a, b, acc[ns]);
      }
    }
  }

  // C/D layout: vgpr r -> row (+8*half), column = lane&15
#pragma unroll
  for (int ns = 0; ns < 8; ++ns) {
    const int col = n0 + ns * 16 + l16;
    const float bc = bias[col];
    float* Cp = C + (size_t)(m0 + wave * 16 + 8 * half) * N + col;
#pragma unroll
    for (int r = 0; r < 8; ++r) Cp[(size_t)r * N] = acc[ns][r] + bc;
  }
}

// Fused multi-query flash attention, fp32 WMMA.
// One wave per (batch, head, 16-row query tile). K/V are the single shared head.
__global__ __launch_bounds__(AT_WAVES * 32) void mqa_flash_wmma(
    const float* __restrict__ Q, const float* __restrict__ Km,
    const float* __restrict__ Vm, float* __restrict__ O) {
  __shared__ float ldsQ[AT_WAVES][16 * HDIM];  // per-wave Q tile (scaled)
  __shared__ float ldsP[AT_WAVES][16 * 16];    // per-wave P staging for re-layout

  const int lane = threadIdx.x & 31;
  const int wave = threadIdx.x >> 5;
  const int half = lane >> 4;
  const int l16  = lane & 15;

  const int tile = blockIdx.x * AT_WAVES + wave;  // over B*H*(S/16)
  const int qt = tile & (SEQ / 16 - 1);
  const int bh = tile / (SEQ / 16);
  const int h  = bh & (HEADS - 1);
  const int b  = bh / HEADS;
  const int q0 = qt << 4;

  const float scale = 0.08838834764831845f;  // 1/sqrt(128)

  float* q_s = ldsQ[wave];
  const float* qg = Q + (size_t)(b * SEQ + q0) * HIDDEN + h * HDIM;
  for (int i = lane; i < 16 * HDIM; i += 32) {
    int r = i >> 7, c = i & (HDIM - 1);
    q_s[i] = qg[(size_t)r * HIDDEN + c] * scale;
  }

  const float* Kb = Km + (size_t)b * SEQ * HDIM;
  const float* Vb = Vm + (size_t)b * SEQ * HDIM;

  float mrow[8], lrow[8];
#pragma unroll
  for (int r = 0; r < 8; ++r) { mrow[r] = -INFINITY; lrow[r] = 0.f; }
  v8f oacc[8] = {};

  float* p_s = ldsP[wave];

  for (int kk0 = 0; kk0 < SEQ; kk0 += 16) {
    // ---- S_tile[16x16] = (Q*scale)[16x128] @ K^T[128x16] ----
    v8f s = {};
    const float* qrow = q_s + l16 * HDIM + 2 * half;
    const float* krow = Kb + (size_t)(kk0 + l16) * HDIM + 2 * half;
#pragma unroll 8
    for (int d = 0; d < HDIM; d += 4) {
      v2f a  = *(const v2f*)(qrow + d);
      v2f bv = *(const v2f*)(krow + d);
      s = wmma_f32_16x16x4(a, bv, s);
    }

    // ---- online softmax (row = 8*half + r; columns striped over 16 lanes) ----
#pragma unroll
    for (int r = 0; r < 8; ++r) {
      float v  = s[r];
      float rm = v;
      rm = fmaxf(rm, __shfl_xor(rm, 1, 32));
      rm = fmaxf(rm, __shfl_xor(rm, 2, 32));
      rm = fmaxf(rm, __shfl_xor(rm, 4, 32));
      rm = fmaxf(rm, __shfl_xor(rm, 8, 32));
      float mnew  = fmaxf(mrow[r], rm);
      float alpha = __expf(mrow[r] - mnew);
      float p     = __expf(v - mnew);
      float rs = p;
      rs += __shfl_xor(rs, 1, 32);
      rs += __shfl_xor(rs, 2, 32);
      rs += __shfl_xor(rs, 4, 32);
      rs += __shfl_xor(rs, 8, 32);
      lrow[r] = lrow[r] * alpha + rs;
      mrow[r] = mnew;
#pragma unroll
      for (int dc = 0; dc < 8; ++dc) oacc[dc][r] *= alpha;
      p_s[(8 * half + r) * 16 + l16] = p;
    }

    // ---- O[16x128] += P[16x16] @ V[16x128] ----
#pragma unroll
    for (int dc = 0; dc < 8; ++dc) {
      const float* vcol = Vb + (size_t)kk0 * HDIM + dc * 16 + l16;
#pragma unroll
      for (int k4 = 0; k4 < 16; k4 += 4) {
        v2f a = *(const v2f*)(p_s + l16 * 16 + k4 + 2 * half);
        v2f bv;
        bv.x = vcol[(size_t)(k4 + 2 * half) * HDIM];
        bv.y = vcol[(size_t)(k4 + 2 * half + 1) * HDIM];
        oacc[dc] = wmma_f32_16x16x4(a, bv, oacc[dc]);
      }
    }
  }

  float* og = O + (size_t)(b * SEQ + q0 + 8 * half) * HIDDEN + h * HDIM + l16;
#pragma unroll
  for (int r = 0; r < 8; ++r) {
    float inv = 1.0f / lrow[r];
#pragma unroll
    for (int dc = 0; dc < 8; ++dc) {
      og[(size_t)r * HIDDEN + dc * 16] = oacc[dc][r] * inv;
    }
  }
}

extern "C" void kernel_launch(void* const* d_in, const int* in_sizes, int n_in,
                              void* d_out, int out_size, void* d_ws, size_t ws_size,
                              hipStream_t stream) {
  const float* x  = (const float*)d_in[0];
  const float* Wq = (const float*)d_in[1];
  const float* bq = (const float*)d_in[2];
  const float* Wk = (const float*)d_in[3];
  const float* bk = (const float*)d_in[4];
  const float* Wv = (const float*)d_in[5];
  const float* bv = (const float*)d_in[6];
  const float* Wo = (const float*)d_in[7];
  const float* bo = (const float*)d_in[8];
  float* out = (float*)d_out;

  constexpr int M = BATCH * SEQ;  // 4096 token rows
  float* q    = (float*)d_ws;                 // [4096, 2048]
  float* k    = q + (size_t)M * HIDDEN;       // [4096, 128]
  float* v    = k + (size_t)M * HDIM;         // [4096, 128]
  float* attn = v + (size_t)M * HDIM;         // [4096, 2048]

  // q = x @ Wq + bq
  gemm_bias_wmma<M, HIDDEN, HIDDEN>
      <<<(M / 128) * (HIDDEN / 128), 256, 0, stream>>>(x, Wq, bq, q);
  // k = x @ Wk + bk ; v = x @ Wv + bv
  gemm_bias_wmma<M, HDIM, HIDDEN>
      <<<(M / 128) * (HDIM / 128), 256, 0, stream>>>(x, Wk, bk, k);
  gemm_bias_wmma<M, HDIM, HIDDEN>
      <<<(M / 128) * (HDIM / 128), 256, 0, stream>>>(x, Wv, bv, v);
  // fused flash attention: one wave per 16-row q tile
  {
    int tiles = BATCH * HEADS * (SEQ / 16);  // 4096
    mqa_flash_wmma<<<tiles / AT_WAVES, AT_WAVES * 32, 0, stream>>>(q, k, v, attn);
  }
  // out = attn @ Wo + bo
  gemm_bias_wmma<M, HIDDEN, HIDDEN>
      <<<(M / 128) * (HIDDEN / 128), 256, 0, stream>>>(attn, Wo, bo, out);
}